// LSTM_84000970375759
// MI455X (gfx1250) — compile-verified
//
#include <hip/hip_runtime.h>
#include <hip/hip_bf16.h>
#include <math.h>

// LSTM single step: V=50257, H=2048, E=1024, L=4, batch=1.
// Memory-bound GEMV chain (~511MB fp32 weights @ 23.3TB/s => ~22us floor).
// Needs ~12MB of loads in flight (BW x latency): 8 waves per 16-row block
// (2 matrices x 4 K-chunks), 4x unrolled WMMA loop (8 b64 loads in flight
// per wave), plus global_prefetch_b8 one cacheline-stream ahead.
// GEMV core: v_wmma_f32_16x16x4_f32; A = 16 weight rows x K=4 (per-lane b64
// matches ISA A layout), B = x broadcast across N (LDS broadcast read).

typedef __attribute__((ext_vector_type(2))) float v2f;
typedef __attribute__((ext_vector_type(4))) float v4f;
typedef __attribute__((ext_vector_type(8))) float v8f;

#define HID   2048
#define EMB   1024
#define G4H   8192   // 4*H

// ---------------------------------------------------------------- embedding
__global__ void lstm_embed_kernel(const int* __restrict__ input,
                                  const float* __restrict__ emb,
                                  float* __restrict__ xbuf) {
    int j = blockIdx.x * blockDim.x + threadIdx.x;
    if (j < EMB) {
        xbuf[j] = emb[(size_t)input[0] * EMB + j];
    }
}

// ------------------------------------------------------------- dual GEMV
// Block = 256 threads (8 waves), covers 16 output rows.
//   waves 0-3: quarter K-chunks of  W1[row,:K1] . x1   (row stride ld1)
//   waves 4-7: quarter K-chunks of  W2[row,:K2] . x2   (row stride ld2)
// out[row] = dot1 + dot2 + b1[row] (+ b2[row] if b2 != nullptr)
// K1, K2 must be multiples of 128 (here: 1024 / 2048).
__global__ __launch_bounds__(256)
void gemv16_dual_wmma(const float* __restrict__ W1, const float* __restrict__ x1,
                      int K1, int ld1,
                      const float* __restrict__ W2, const float* __restrict__ x2,
                      int K2, int ld2,
                      const float* __restrict__ b1, const float* __restrict__ b2,
                      float* __restrict__ out) {
    __shared__ float xs[2][HID];   // staged x per matrix (<= 8KB each)
    __shared__ float red[8][16];   // per-wave partial sums for 16 rows

    const int tid  = threadIdx.x;
    const int wave = tid >> 5;     // 0..7
    const int grp  = wave >> 2;    // 0 -> matrix 1, 1 -> matrix 2
    const int sub  = wave & 3;     // K-quarter within the matrix
    const int lane = tid & 31;
    const int m    = lane & 15;    // matrix row within 16-row tile
    const int half = lane >> 4;    // 0: K=0,1  1: K=2,3 (ISA A/B f32 layout)
    const int row0 = blockIdx.x * 16;

    const float* W  = grp ? W2 : W1;
    const float* xg = grp ? x2 : x1;
    const int    K  = grp ? K2 : K1;
    const int    ld = grp ? ld2 : ld1;
    const int    Kc   = K >> 2;        // per-wave chunk (multiple of 32)
    const int    koff = sub * Kc;

    // each wave stages its own x chunk into LDS (wave-local region)
    for (int k = lane * 4; k < Kc; k += 128) {
        *(v4f*)&xs[grp][koff + k] = *(const v4f*)&xg[koff + k];
    }
    __syncthreads();

    // per-lane base pointers matching the WMMA fp32 A/B VGPR layout:
    //   lanes 0-15 load (K0,K1) of row m, lanes 16-31 load (K2,K3) of row m
    const float* Wrow = W + (size_t)(row0 + m) * ld + koff + 2 * half;
    const float* xl   = &xs[grp][koff + 2 * half];

    v8f c0 = {};
    v8f c1 = {};
    for (int k0 = 0; k0 < Kc; k0 += 32) {
        // one 128B cacheline per row consumed per iteration; prefetch the
        // line needed 4 iterations from now (global_prefetch_b8, no LOADcnt)
        __builtin_prefetch(Wrow + k0 + 128, 0, 3);
        #pragma unroll
        for (int u = 0; u < 4; ++u) {
            const int kk = k0 + u * 8;
            v2f a0  = *(const v2f*)(Wrow + kk);       // global_load_b64
            v2f bb0 = *(const v2f*)(xl + kk);         // ds_load broadcast
            v2f a1  = *(const v2f*)(Wrow + kk + 4);
            v2f bb1 = *(const v2f*)(xl + kk + 4);
            // D = A(16x4 weights) x B(4x16 broadcast x) + C, full fp32
            c0 = __builtin_amdgcn_wmma_f32_16x16x4_f32(false, a0, false, bb0,
                                                       (short)0, c0, false, false);
            c1 = __builtin_amdgcn_wmma_f32_16x16x4_f32(false, a1, false, bb1,
                                                       (short)0, c1, false, false);
        }
    }
    v8f c = c0 + c1;  // identical D layout -> elementwise add is matrix add

    // D[m][n] identical for all n. Lane 0 holds rows 0..7 in VGPRs 0..7,
    // lane 16 holds rows 8..15. Dump 16 partial sums per wave to LDS.
    if (m == 0) {
        #pragma unroll
        for (int v = 0; v < 8; ++v) {
            red[wave][half * 8 + v] = c[v];
        }
    }
    __syncthreads();

    if (tid < 16) {
        float acc = 0.0f;
        #pragma unroll
        for (int w = 0; w < 8; ++w) acc += red[w][tid];
        acc += b1[row0 + tid];
        if (b2) acc += b2[row0 + tid];
        out[row0 + tid] = acc;
    }
}

// --------------------------------------------------------- LSTM cell update
// gates layout: [i | f | g | o], each H wide. Writes h,c to d_out slots and
// h also to xbuf (input of next layer).
__global__ void lstm_cell_kernel(const float* __restrict__ gates,
                                 const float* __restrict__ c_in,
                                 float* __restrict__ h_out,
                                 float* __restrict__ c_out,
                                 float* __restrict__ xbuf) {
    int j = blockIdx.x * blockDim.x + threadIdx.x;
    if (j >= HID) return;
    float ig = gates[j];
    float fg = gates[HID + j];
    float gg = gates[2 * HID + j];
    float og = gates[3 * HID + j];
    float i = 1.0f / (1.0f + expf(-ig));
    float f = 1.0f / (1.0f + expf(-fg));
    float g = tanhf(gg);
    float o = 1.0f / (1.0f + expf(-og));
    float c = f * c_in[j] + i * g;
    float h = o * tanhf(c);
    h_out[j] = h;
    c_out[j] = c;
    xbuf[j] = h;
}

// ------------------------------------------------------------------- launch
extern "C" void kernel_launch(void* const* d_in, const int* in_sizes, int n_in,
                              void* d_out, int out_size, void* d_ws, size_t ws_size,
                              hipStream_t stream) {
    const int*   input  = (const int*)d_in[0];
    const float* hidden = (const float*)d_in[1];   // [4, 2048]
    const float* cell   = (const float*)d_in[2];   // [4, 2048]
    const float* emb    = (const float*)d_in[3];   // [V, 1024]
    const float* W_ih0  = (const float*)d_in[4];   // [8192, 1024]
    const float* W_ihR  = (const float*)d_in[5];   // [3, 8192, 2048]
    const float* W_hh   = (const float*)d_in[6];   // [4, 8192, 2048]
    const float* b_ih   = (const float*)d_in[7];   // [4, 8192]
    const float* b_hh   = (const float*)d_in[8];   // [4, 8192]
    const float* fc_W   = (const float*)d_in[9];   // [1024, 2048]
    const float* fc_b   = (const float*)d_in[10];  // [1024]
    float* out = (float*)d_out;                    // [1024 | 4*2048 | 4*2048]

    float* xbuf  = (float*)d_ws;        // 2048 floats: current layer input
    float* gates = xbuf + HID;          // 8192 floats: gate pre-activations

    lstm_embed_kernel<<<(EMB + 255) / 256, 256, 0, stream>>>(input, emb, xbuf);

    for (int l = 0; l < 4; ++l) {
        const float* Wih = (l == 0) ? W_ih0
                                    : (W_ihR + (size_t)(l - 1) * G4H * HID);
        const int Kin = (l == 0) ? EMB : HID;
        const float* Whh = W_hh + (size_t)l * G4H * HID;

        gemv16_dual_wmma<<<G4H / 16, 256, 0, stream>>>(
            Wih, xbuf, Kin, Kin,
            Whh, hidden + l * HID, HID, HID,
            b_ih + l * G4H, b_hh + l * G4H, gates);

        lstm_cell_kernel<<<(HID + 255) / 256, 256, 0, stream>>>(
            gates, cell + l * HID,
            out + EMB + l * HID,             // hidden_new[l]
            out + EMB + 4 * HID + l * HID,   // cell_new[l]
            xbuf);
    }

    // decoded = fc_W @ h3 + fc_b ; K=2048 split across the two wave groups
    gemv16_dual_wmma<<<EMB / 16, 256, 0, stream>>>(
        fc_W, xbuf, 1024, HID,
        fc_W + 1024, xbuf + 1024, 1024, HID,
        fc_b, nullptr, out);
}